// NeuronGraph_39238821216886
// MI455X (gfx1250) — compile-verified
//
#include <hip/hip_runtime.h>
#include <math.h>

// ---- Problem constants (match reference) ----
#define N_IN     10000
#define HID      30000
#define N_OUT    1000
#define N_TOTAL  131000   // total neurons
#define N_NONIN  121000   // non-input neurons
#define FAN      128
#define E_REC    4000000
#define MAX_PREV 30000    // largest "previous level" (fits in 120KB LDS)

// 128-bit vector type matching the async-to-LDS builtin's expected pointee
typedef int v4i __attribute__((vector_size(16)));

// Relaxed agent-scope float atomic add -> global_atomic_add_f32 (no return)
__device__ __forceinline__ void atomicAddRelaxed(float* p, float v) {
  __hip_atomic_fetch_add(p, v, __ATOMIC_RELAXED, __HIP_MEMORY_SCOPE_AGENT);
}

// acc <- bias ; cur[0:N_IN] <- obs
__global__ void ng_init_kernel(const float* __restrict__ obs,
                               const float* __restrict__ bias,
                               float* __restrict__ cur,
                               float* __restrict__ acc) {
  int i = blockIdx.x * blockDim.x + threadIdx.x;
  if (i < N_IN)    cur[i] = obs[i];
  if (i < N_NONIN) acc[i] = bias[i];
}

// acc[dst[e]] += w[e] * hist[src[e]] over 4M delay-1 edges.
// Edge streams are consumed 4-at-a-time with b128 loads; hist is L2-resident.
__global__ void ng_rec_scatter_kernel(const float* __restrict__ hist,
                                      const int*   __restrict__ src,
                                      const int*   __restrict__ dst,
                                      const float* __restrict__ w,
                                      int nE4,   // nE/4
                                      float* __restrict__ acc) {
  const int4*   src4 = (const int4*)src;
  const int4*   dst4 = (const int4*)dst;
  const float4* w4   = (const float4*)w;
  int stride = gridDim.x * blockDim.x;
  for (int i = blockIdx.x * blockDim.x + threadIdx.x; i < nE4; i += stride) {
    __builtin_prefetch(&src4[i + 2 * stride], 0, 0);
    __builtin_prefetch(&dst4[i + 2 * stride], 0, 0);
    __builtin_prefetch(&w4[i + 2 * stride], 0, 0);
    int4   s = src4[i];
    int4   d = dst4[i];
    float4 ww = w4[i];
    atomicAddRelaxed(&acc[d.x], ww.x * hist[s.x]);
    atomicAddRelaxed(&acc[d.y], ww.y * hist[s.y]);
    atomicAddRelaxed(&acc[d.z], ww.z * hist[s.z]);
    atomicAddRelaxed(&acc[d.w], ww.w * hist[s.w]);
  }
}

// Feedforward scatter for one level: stage the ENTIRE previous-level activation
// slice (<=120KB) into LDS with gfx1250 async global->LDS DMA, then gather from
// LDS and atomic-scatter into acc (level-local).
__global__ void ng_ff_scatter_lds_kernel(const float* __restrict__ curPrev, // cur + prevOff (16B aligned)
                                         int prevSz, int prevOff,
                                         const int*   __restrict__ src,     // global indices
                                         const int*   __restrict__ dst,     // level-local
                                         const float* __restrict__ w,
                                         int nE4,                           // nE/4
                                         float* __restrict__ acc) {         // acc + pos
  __shared__ __align__(16) float s_prev[MAX_PREV];
  const int tid = threadIdx.x;
  const int nChunks = prevSz >> 2;  // prevSz is always a multiple of 4

#if defined(__gfx1250__) && __has_builtin(__builtin_amdgcn_global_load_async_to_lds_b128)
  {
    typedef __attribute__((address_space(3))) v4i as3_v4i;
    for (int i = tid; i < nChunks; i += blockDim.x) {
      __builtin_amdgcn_global_load_async_to_lds_b128(
          (v4i*)(curPrev + 4 * i),
          (as3_v4i*)(s_prev + 4 * i),
          /*offset=*/0, /*cpol=*/0);
    }
#if __has_builtin(__builtin_amdgcn_s_wait_asynccnt)
    __builtin_amdgcn_s_wait_asynccnt(0);
#else
    asm volatile("s_wait_asynccnt 0" ::: "memory");
#endif
  }
#else
  {
    const float4* g4 = (const float4*)curPrev;
    float4* s4 = (float4*)s_prev;
    for (int i = tid; i < nChunks; i += blockDim.x) s4[i] = g4[i];
  }
#endif
  __syncthreads();

  const int4*   src4 = (const int4*)src;
  const int4*   dst4 = (const int4*)dst;
  const float4* w4   = (const float4*)w;
  const int stride = gridDim.x * blockDim.x;
  for (int i = blockIdx.x * blockDim.x + tid; i < nE4; i += stride) {
    __builtin_prefetch(&src4[i + 2 * stride], 0, 0);
    __builtin_prefetch(&dst4[i + 2 * stride], 0, 0);
    __builtin_prefetch(&w4[i + 2 * stride], 0, 0);
    int4   s = src4[i];
    int4   d = dst4[i];
    float4 ww = w4[i];
    atomicAddRelaxed(&acc[d.x], ww.x * s_prev[s.x - prevOff]);
    atomicAddRelaxed(&acc[d.y], ww.y * s_prev[s.y - prevOff]);
    atomicAddRelaxed(&acc[d.z], ww.z * s_prev[s.z - prevOff]);
    atomicAddRelaxed(&acc[d.w], ww.w * s_prev[s.w - prevOff]);
  }
}

// Direct (no-LDS) variant for the tiny output level (128K edges)
__global__ void ng_ff_scatter_direct_kernel(const float* __restrict__ cur,
                                            const int*   __restrict__ src,
                                            const int*   __restrict__ dst,
                                            const float* __restrict__ w,
                                            int nE4,
                                            float* __restrict__ acc) {
  const int4*   src4 = (const int4*)src;
  const int4*   dst4 = (const int4*)dst;
  const float4* w4   = (const float4*)w;
  int stride = gridDim.x * blockDim.x;
  for (int i = blockIdx.x * blockDim.x + threadIdx.x; i < nE4; i += stride) {
    int4   s = src4[i];
    int4   d = dst4[i];
    float4 ww = w4[i];
    atomicAddRelaxed(&acc[d.x], ww.x * cur[s.x]);
    atomicAddRelaxed(&acc[d.y], ww.y * cur[s.y]);
    atomicAddRelaxed(&acc[d.z], ww.z * cur[s.z]);
    atomicAddRelaxed(&acc[d.w], ww.w * cur[s.w]);
  }
}

// cur[OFF[l]+i] = tanh(acc[pos+i]); level 5 also writes d_out
__global__ void ng_tanh_kernel(const float* __restrict__ accLevel,
                               float* __restrict__ curLevel,
                               float* __restrict__ out, int n) {
  int i = blockIdx.x * blockDim.x + threadIdx.x;
  if (i < n) {
    float v = tanhf(accLevel[i]);
    curLevel[i] = v;
    if (out) out[i] = v;
  }
}

extern "C" void kernel_launch(void* const* d_in, const int* in_sizes, int n_in,
                              void* d_out, int out_size, void* d_ws, size_t ws_size,
                              hipStream_t stream) {
  (void)in_sizes; (void)n_in; (void)out_size; (void)ws_size;

  const float* obs  = (const float*)d_in[0];
  const float* hist = (const float*)d_in[1];
  const float* bias = (const float*)d_in[2];
  const int*   ff_src[5];
  const int*   ff_dst[5];
  const float* ff_w[5];
  for (int l = 0; l < 5; ++l) {
    ff_src[l] = (const int*)  d_in[3 + 3 * l];
    ff_dst[l] = (const int*)  d_in[4 + 3 * l];
    ff_w[l]   = (const float*)d_in[5 + 3 * l];
  }
  const int*   rec_src = (const int*)  d_in[18];
  const int*   rec_dst = (const int*)  d_in[19];
  const float* rec_w   = (const float*)d_in[20];

  // Workspace layout: cur[N_TOTAL] | acc[N_NONIN]  (~1 MB total, 16B aligned)
  float* cur = (float*)d_ws;
  float* acc = cur + N_TOTAL;

  static const int OFF[7] = {0, 10000, 40000, 70000, 100000, 130000, 131000};
  static const int SZ[6]  = {N_IN, HID, HID, HID, HID, N_OUT};

  // 1) init: acc <- bias, cur[:N_IN] <- obs
  ng_init_kernel<<<(N_NONIN + 255) / 256, 256, 0, stream>>>(obs, bias, cur, acc);

  // 2) recurrent edges (delay=1): independent of topo order
  ng_rec_scatter_kernel<<<2048, 256, 0, stream>>>(hist, rec_src, rec_dst, rec_w,
                                                  E_REC / 4, acc);

  // 3) levels 1..5 in topological order (stream-ordered launches)
  int pos = 0;
  for (int l = 1; l <= 5; ++l) {
    const int sz      = SZ[l];
    const int nE      = sz * FAN;
    const int prevOff = OFF[l - 1];
    const int prevSz  = SZ[l - 1];

    if (l < 5) {
      // 120KB LDS staging of previous level -> 2 workgroups / WGP
      ng_ff_scatter_lds_kernel<<<256, 256, 0, stream>>>(
          cur + prevOff, prevSz, prevOff,
          ff_src[l - 1], ff_dst[l - 1], ff_w[l - 1], nE / 4, acc + pos);
    } else {
      ng_ff_scatter_direct_kernel<<<125, 256, 0, stream>>>(
          cur, ff_src[l - 1], ff_dst[l - 1], ff_w[l - 1], nE / 4, acc + pos);
    }

    float* outPtr = (l == 5) ? (float*)d_out : (float*)nullptr;
    ng_tanh_kernel<<<(sz + 255) / 256, 256, 0, stream>>>(acc + pos, cur + OFF[l],
                                                         outPtr, sz);
    pos += sz;
  }
}